// FeaturesFactorizationModule_3693671875229
// MI455X (gfx1250) — compile-verified
//
#include <hip/hip_runtime.h>

// MI455X / gfx1250 — wave32, WMMA, TDM, 320KB LDS/WGP, 192MB L2, 23.3 TB/s HBM.
// Rank-1 NMF (R=1, STEPS=4) between 1x1x1 convs. Memory-bound: ~470MB HBM
// traffic -> ~20us floor. h (94MB) stays L2-resident for all NMF passes.

typedef float v2f __attribute__((ext_vector_type(2)));
typedef float v8f __attribute__((ext_vector_type(8)));
typedef unsigned int u32x4 __attribute__((ext_vector_type(4)));
typedef int i32x4 __attribute__((ext_vector_type(4)));
typedef int i32x8 __attribute__((ext_vector_type(8)));

constexpr int   NB   = 8;        // batch
constexpr int   NHW  = 2304;     // 48*48
constexpr int   ND   = 160;      // D = T
constexpr int   NN   = 18432;    // N = 8*48*48
constexpr int   NG   = 20;       // T/8
constexpr long  THW  = 368640L;  // 160*2304 (one channel-plane)
constexpr long  DN   = 2949120L; // 160*18432 per-batch h block
constexpr long  XB   = 5898240L; // per-batch x / out stride (16 channels)
constexpr float FEPS = 1e-6f;

#ifndef __has_builtin
#define __has_builtin(x) 0
#endif
#if defined(__HIP_DEVICE_COMPILE__) && __has_builtin(__builtin_amdgcn_tensor_load_to_lds)
#define USE_TDM 1
#else
#define USE_TDM 0
#endif

// ---------------- init: coef=1 (softmax of size-1 axis), bases=1/sqrt(160),
// btb=||b0||^2=1, zero accumulators ----------------
__global__ __launch_bounds__(256) void k_init(float* coef, float* bases,
                                              float* num_d, float* btb, float* ctc) {
  int i = blockIdx.x * 256 + threadIdx.x;
  if (i < NB * NN) coef[i] = 1.0f;
  if (i < NB * ND) { bases[i] = 0.07905694150420949f; num_d[i] = 0.0f; }
  if (i < NB)      { btb[i] = 1.0f; ctc[i] = 0.0f; }
}

// ---------------- pre: h = relu(W_pre @ x), 16->8 channels ----------------
__global__ __launch_bounds__(256) void k_pre(const float* __restrict__ x,
                                             const float* __restrict__ Wp,
                                             float* __restrict__ h) {
  __shared__ float w[128];
  int tid = threadIdx.x;
  if (tid < 128) w[tid] = Wp[tid];
  __syncthreads();
  long p   = (long)blockIdx.x * 256 + tid;   // 0 .. 8*368640-1 (exact grid)
  int  b   = (int)(p / THW);
  int  rem = (int)(p % THW);
  const float* xb = x + (size_t)b * XB + rem;
  float acc[8] = {};
#pragma unroll
  for (int c = 0; c < 16; ++c) {
    float xv = xb[(size_t)c * THW];
#pragma unroll
    for (int o = 0; o < 8; ++o) acc[o] = fmaf(w[o * 16 + c], xv, acc[o]);
  }
  float* hb = h + (size_t)b * DN + rem;
#pragma unroll
  for (int o = 0; o < 8; ++o) hb[(size_t)o * THW] = fmaxf(acc[o], 0.0f);
}

// ---------------- one fused NMF iteration: TDM-stage a 160x64 column tile of
// h into LDS (one TENSOR_LOAD_TO_LDS per block), then num_n -> coef update ->
// num_d & ctc accumulation, all from the LDS tile. ----------------
__global__ __launch_bounds__(256) void k_iter(const float* __restrict__ h,
                                              float* __restrict__ coef,
                                              const float* __restrict__ bases,
                                              const float* __restrict__ btb,
                                              float* __restrict__ num_d,
                                              float* __restrict__ ctc) {
  __shared__ float tile[ND * 64];   // contiguous 160x64 tile (TDM row-major)
  __shared__ float bs[ND];
  __shared__ float red[256];
  __shared__ float cfs[64];
  __shared__ float btbsh;
  int tid = threadIdx.x;
  int b   = blockIdx.y;
  int n0  = blockIdx.x * 64;
  const float* hb = h + (size_t)b * DN;

  if (tid < ND) bs[tid] = bases[b * ND + tid];
  if (tid == 0) btbsh = btb[b];

#if USE_TDM
  if (tid < 32) {  // wave 0 issues one TDM DMA for the whole 40KB tile
    unsigned lds_base = (unsigned)(unsigned long long)(void*)&tile[0];
    unsigned long long ga = (unsigned long long)(const void*)(hb + n0);
    u32x4 g0;
    g0[0] = 1u;                                   // count=1 user descriptor
    g0[1] = lds_base;                             // lds_addr (bytes)
    g0[2] = (unsigned)(ga & 0xffffffffu);         // global_addr[31:0]
    g0[3] = (unsigned)((ga >> 32) & 0x1ffffffu)   // global_addr[56:32]
            | 0x80000000u;                        // type=2 ("image")
    i32x8 g1;
    g1[0] = 0x00020000;                // data_size=2 (4B); no multicast/pad
    g1[1] = (int)(18432u << 16);       // tensor_dim0=18432 (low16 in [31:16])
    g1[2] = (int)(160u << 16);         // dim0 hi16=0 | tensor_dim1=160 low16
    g1[3] = (int)(64u << 16);          // tensor_dim1 hi16=0 | tile_dim0=64
    g1[4] = 160;                       // tile_dim1=160, tile_dim2=0
    g1[5] = 18432;                     // tensor_dim0_stride low32
    g1[6] = 0;                         // stride hi16 | tensor_dim1_stride low16
    g1[7] = 0;                         // tensor_dim1_stride hi32 (unused, 2D)
    i32x4 gz = {0, 0, 0, 0};           // groups 2/3: dims 3/4 unused
#if defined(__clang_major__) && (__clang_major__ >= 23)
    i32x8 gz8 = {0, 0, 0, 0, 0, 0, 0, 0};
    __builtin_amdgcn_tensor_load_to_lds(g0, g1, gz, gz, gz8, 0);
#else
    __builtin_amdgcn_tensor_load_to_lds(g0, g1, gz, gz, 0);
#endif
    __builtin_amdgcn_s_wait_tensorcnt(0);
  }
#else
  for (int k = tid; k < ND * 16; k += 256) {      // fallback: 160 rows x 16 f4
    int d = k >> 4, q = k & 15;
    float4 v = *(const float4*)(hb + (size_t)d * NN + n0 + q * 4);
    float* t = &tile[d * 64 + q * 4];
    t[0] = v.x; t[1] = v.y; t[2] = v.z; t[3] = v.w;
  }
#endif
  __syncthreads();

  int grp = tid >> 6, nn = tid & 63;              // 4 groups split the d-range
  float p = 0.0f;
  int d0 = grp * 40;
  for (int d = d0; d < d0 + 40; ++d) p = fmaf(tile[d * 64 + nn], bs[d], p);
  red[tid] = p;
  __syncthreads();

  if (grp == 0) {
    float numv = red[nn] + red[64 + nn] + red[128 + nn] + red[192 + nn];
    float co   = coef[(size_t)b * NN + n0 + nn];
    float cn   = co * numv / fmaf(co, btbsh, FEPS);
    coef[(size_t)b * NN + n0 + nn] = cn;
    cfs[nn] = cn;
    red[nn] = cn * cn;
  }
  __syncthreads();

  if (tid == 0) {
    float s = 0.0f;
    for (int i = 0; i < 64; ++i) s += red[i];
    atomicAdd(&ctc[b], s);
  }
  if (tid < ND) {                     // rotation => 32 distinct banks per wave
    float acc = 0.0f;
    int rot = tid & 63;
#pragma unroll 4
    for (int j = 0; j < 64; ++j) {
      int c = (j + rot) & 63;
      acc = fmaf(tile[tid * 64 + c], cfs[c], acc);
    }
    atomicAdd(&num_d[b * ND + tid], acc);
  }
}

// ---------------- tiny: bases update, new btb, reset accumulators ----------
__global__ __launch_bounds__(256) void k_bases(float* bases, float* btb,
                                               float* num_d, float* ctc) {
  __shared__ float bt[8];
  int tid = threadIdx.x;
  if (tid < 8) bt[tid] = 0.0f;
  __syncthreads();
  for (int i = tid; i < NB * ND; i += 256) {
    int b = i / ND;
    float bo = bases[i];
    float bn = bo * num_d[i] / fmaf(bo, ctc[b], FEPS);
    bases[i] = bn;
    num_d[i] = 0.0f;
    atomicAdd(&bt[b], bn * bn);
  }
  __syncthreads();
  if (tid < 8) { btb[tid] = bt[tid]; ctc[tid] = 0.0f; }
}

// ---------------- tiny: M[b][g][oc] = W2 @ relu(W1 @ bvec(g)); collapses the
// whole post-conv via ReLU positive homogeneity (coef >= 0) ----------------
__global__ __launch_bounds__(256) void k_postm(const float* __restrict__ bases,
                                               const float* __restrict__ W1,
                                               const float* __restrict__ W2,
                                               float* __restrict__ M) {
  int tid = threadIdx.x;
  for (int e = tid; e < NB * NG * 16; e += 256) {
    int b = e / (NG * 16);
    int r = e % (NG * 16);
    int g = r / 16, oc = r % 16;
    float acc = 0.0f;
#pragma unroll
    for (int j = 0; j < 8; ++j) {
      float y1 = 0.0f;
#pragma unroll
      for (int i = 0; i < 8; ++i)
        y1 = fmaf(W1[j * 8 + i], bases[b * ND + i * NG + g], y1);
      y1  = fmaxf(y1, 0.0f);
      acc = fmaf(W2[oc * 8 + j], y1, acc);
    }
    M[e] = acc;
  }
}

// ---------------- final: coef update with b4, then y = coef[n] * M[g] via
// V_WMMA_F32_16X16X4_F32 rank-1 outer products (A col0 = M[g], B row0 = coef).
// D layout: VGPR r, lane l -> row oc = r + 8*(l/16), col = point l%16. -------
__global__ __launch_bounds__(256) void k_out(const float* __restrict__ h,
                                             const float* __restrict__ coef,
                                             const float* __restrict__ bases,
                                             const float* __restrict__ btb,
                                             const float* __restrict__ M,
                                             float* __restrict__ out) {
  __shared__ float bs[ND];
  __shared__ float Ms[NG * 16];
  __shared__ float btbsh;
  int tid = threadIdx.x;
  int b   = blockIdx.y;
  if (tid < ND) bs[tid] = bases[b * ND + tid];
  for (int i = tid; i < NG * 16; i += 256) Ms[i] = M[b * NG * 16 + i];
  if (tid == 0) btbsh = btb[b];
  __syncthreads();

  int w = tid >> 5, lane = tid & 31;
  int n0 = blockIdx.x * 128 + w * 16;     // 16 points per wave, one t8-slab
  int n  = n0 + (lane & 15);
  const float* hb = h + (size_t)b * DN;

  // num_n: lanes 0-15 do d=[0,80), lanes 16-31 do d=[80,160); swap-add halves
  float num = 0.0f;
  int dstart = (lane >> 4) * 80;
  for (int d = dstart; d < dstart + 80; ++d)
    num = fmaf(hb[(size_t)d * NN + n], bs[d], num);
  num += __shfl_xor(num, 16, 32);

  float co = coef[(size_t)b * NN + n];
  float cf = co * num / fmaf(co, btbsh, FEPS);

  int t8   = n0 / NHW;
  int hw   = n0 % NHW + (lane & 15);
  int half = lane >> 4;
  v2f bv = { half ? 0.0f : cf, 0.0f };    // B row0 = coef tile, rest 0
  size_t base = (size_t)b * XB + (size_t)(half * 8) * THW + (size_t)t8 * NHW + hw;

#pragma unroll
  for (int g = 0; g < NG; ++g) {
    float msv = Ms[g * 16 + (lane & 15)];   // uniform LDS load, no EXEC dance
    v2f a = { half ? 0.0f : msv, 0.0f };    // A col0 = M[g], rest 0
    v8f c = {};
    c = __builtin_amdgcn_wmma_f32_16x16x4_f32(false, a, false, bv,
                                              (short)0, c, false, false);
    float* op = out + base + (size_t)(g * 8) * NHW;            // t = g*8 + t8
#pragma unroll
    for (int r = 0; r < 8; ++r) op[(size_t)r * THW] = c[r];    // oc = r + 8*half
  }
}

extern "C" void kernel_launch(void* const* d_in, const int* in_sizes, int n_in,
                              void* d_out, int out_size, void* d_ws, size_t ws_size,
                              hipStream_t stream) {
  const float* x  = (const float*)d_in[0];
  const float* Wp = (const float*)d_in[1];
  const float* W1 = (const float*)d_in[2];
  const float* W2 = (const float*)d_in[3];
  float* out = (float*)d_out;

  float* ws    = (float*)d_ws;              // ~95 MB of fp32 scratch
  float* h     = ws;                        // 8*160*18432
  float* coef  = h     + (size_t)NB * DN;   // 8*18432
  float* bases = coef  + (size_t)NB * NN;   // 8*160
  float* num_d = bases + NB * ND;           // 8*160
  float* btb   = num_d + NB * ND;           // 8
  float* ctc   = btb   + NB;                // 8
  float* M     = ctc   + NB;                // 8*20*16

  k_init<<<(NB * NN + 255) / 256, 256, 0, stream>>>(coef, bases, num_d, btb, ctc);
  k_pre<<<(int)((NB * THW) / 256), 256, 0, stream>>>(x, Wp, h);
  dim3 gi(NN / 64, NB);
  for (int s = 0; s < 4; ++s) {
    k_iter<<<gi, 256, 0, stream>>>(h, coef, bases, btb, num_d, ctc);
    k_bases<<<1, 256, 0, stream>>>(bases, btb, num_d, ctc);
  }
  k_postm<<<1, 256, 0, stream>>>(bases, W1, W2, M);
  dim3 go(NN / 128, NB);
  k_out<<<go, 256, 0, stream>>>(h, coef, bases, btb, M, out);
}